// SimpleSSMLayer_35828617183812
// MI455X (gfx1250) — compile-verified
//
#include <hip/hip_runtime.h>

// Two-pass segmented SSM scan for MI455X (gfx1250, wave32).
//   h[l] = e^{-a} h[l-1] + B x[l],  y[l,d] = sum_n C*f*h + x*D,
//   f = 1/(1+eps*e^{-(l+1)a}) tracked recursively (g *= e^{-a}).
// Pass 1: per-segment zero-start end states E_j (carry pass).
// Pass 2: h0 = sum_{j<s} E_j * e^{-a*L*(s-1-j)}, then chunked scan;
//         per 16-step chunk the sum over 16 states is 4 chained
//         v_wmma_f32_16x16x4_f32 (B = ones). x staged to LDS with
//         double-buffered global_load_async_to_lds_b32 when available.

typedef float v2f __attribute__((ext_vector_type(2)));
typedef float v8f __attribute__((ext_vector_type(8)));

#ifndef __has_builtin
#define __has_builtin(x) 0
#endif
#if __has_builtin(__builtin_amdgcn_wmma_f32_16x16x4_f32)
#define USE_WMMA_F32 1
#else
#define USE_WMMA_F32 0
#endif
#if __has_builtin(__builtin_amdgcn_global_load_async_to_lds_b32) && \
    __has_builtin(__builtin_amdgcn_s_wait_asynccnt)
#define USE_ASYNC_LDS 1
#else
#define USE_ASYNC_LDS 0
#endif

#if USE_ASYNC_LDS
using as1_i32 = __attribute__((address_space(1))) int;   // global
using as3_i32 = __attribute__((address_space(3))) int;   // LDS
#endif

namespace {
constexpr int BATCH = 4;
constexpr int SEQ   = 2048;
constexpr int DIM   = 512;
constexpr int STATE = 16;
constexpr float EPS = 1e-6f;

constexpr int CHUNK        = 16;                  // time steps per WMMA tile
constexpr int WAVES        = 8;                   // waves per block
constexpr int CH_PER_WAVE  = 2;                   // 2 channels x 16 states
constexpr int CH_PER_BLOCK = WAVES * CH_PER_WAVE; // 16
constexpr int GROUPS       = DIM / CH_PER_BLOCK;  // 32
constexpr int BLOCKS_PER_SEG = BATCH * GROUPS;    // 128
constexpr int NSEG   = 8;
constexpr int SEGLEN = SEQ / NSEG;                // 256
constexpr float CLAMPV = 1e30f;
}

__device__ __forceinline__ float clampf(float h) {
    return __builtin_amdgcn_fmed3f(h, -CLAMPV, CLAMPV);   // single v_med3_f32
}

// ---------------- Pass 1: segment carry states --------------------------
__global__ __launch_bounds__(WAVES * 32)
void ssm_carry_kernel(const float* __restrict__ x,
                      const float* __restrict__ A,
                      const float* __restrict__ B,
                      float* __restrict__ E)
{
    const int lane = threadIdx.x & 31;
    const int w    = threadIdx.x >> 5;
    const int half = lane >> 4;
    const int n    = lane & 15;

    const int seg   = blockIdx.x / BLOCKS_PER_SEG;    // 0 .. NSEG-2
    const int rem   = blockIdx.x % BLOCKS_PER_SEG;
    const int batch = rem / GROUPS;
    const int ch    = (rem % GROUPS) * CH_PER_BLOCK + w * CH_PER_WAVE + half;

    const float a  = A[ch * STATE + n];
    const float Bv = B[ch * STATE + n];
    const float r  = __expf(-a);

    const float* xp = x + ((size_t)batch * SEQ + (size_t)seg * SEGLEN) * DIM + ch;
    float h = 0.0f;
    #pragma unroll 8
    for (int l = 0; l < SEGLEN; ++l) {
        float xt = xp[(size_t)l * DIM];
        if (l + 8 < SEGLEN) __builtin_prefetch(&xp[(size_t)(l + 8) * DIM], 0, 0);
        h = clampf(__builtin_fmaf(r, h, Bv * xt));
    }
    E[(((size_t)seg * BATCH + batch) * DIM + ch) * STATE + n] = h;
}

// ---------------- Pass 2: outputs ---------------------------------------
__global__ __launch_bounds__(WAVES * 32)
void ssm_out_kernel(const float* __restrict__ x,
                    const float* __restrict__ A,
                    const float* __restrict__ B,
                    const float* __restrict__ C,
                    const float* __restrict__ Dv,
                    const float* __restrict__ E,
                    float* __restrict__ out,
                    int seglen)
{
    __shared__ float Zbuf[WAVES][CH_PER_WAVE][STATE][CHUNK + 1]; // [n][t] padded
    __shared__ float Xbuf[WAVES][2][CH_PER_WAVE][CHUNK];         // double buffered

    const int lane = threadIdx.x & 31;
    const int w    = threadIdx.x >> 5;
    const int half = lane >> 4;
    const int n    = lane & 15;

    const int seg    = blockIdx.x / BLOCKS_PER_SEG;
    const int rem    = blockIdx.x % BLOCKS_PER_SEG;
    const int batch  = rem / GROUPS;
    const int dblock = (rem % GROUPS) * CH_PER_BLOCK;
    const int ch0    = dblock + w * CH_PER_WAVE;
    const int ch     = ch0 + half;
    const int l0base = seg * seglen;

    const float a   = A[ch * STATE + n];
    const float Bv  = B[ch * STATE + n];
    const float Cv  = C[ch * STATE + n];
    const float Dd0 = Dv[ch0];
    const float Dd1 = Dv[ch0 + 1];
    const float r   = __expf(-a);

    // Reconstruct segment-start state from carries: weight r^(L*(seg-1-j)).
    float h = 0.0f;
    for (int j = 0; j < seg; ++j) {
        float wgt = __expf(-a * (float)(seglen * (seg - 1 - j)));
        float Ej  = E[(((size_t)j * BATCH + batch) * DIM + ch) * STATE + n];
        h = __builtin_fmaf(Ej, wgt, h);
    }
    h = clampf(h);
    float g = EPS * __expf(-a * (float)l0base);  // g_l = eps*r^(l+1) via g*=r

    const float* xb = x   + (size_t)batch * SEQ * DIM;
    float*       ob = out + (size_t)batch * SEQ * DIM;
    const int nchunks = seglen / CHUNK;

    auto stage = [&](int c, int buf) {
        const float* gp = &xb[(size_t)(l0base + c * CHUNK + n) * DIM + ch];
        float*       lp = &Xbuf[w][buf][half][n];
#if USE_ASYNC_LDS
        __builtin_amdgcn_global_load_async_to_lds_b32(
            (as1_i32*)gp, (as3_i32*)lp, 0, 0);
#else
        *lp = *gp;
#endif
    };

    int pb = 0;
    stage(0, pb);
    for (int c = 0; c < nchunks; ++c) {
        if (c + 1 < nchunks) stage(c + 1, pb ^ 1);
#if USE_ASYNC_LDS
        if (c + 1 < nchunks) __builtin_amdgcn_s_wait_asynccnt(1);
        else                 __builtin_amdgcn_s_wait_asynccnt(0);
#endif
        __builtin_amdgcn_wave_barrier();
        asm volatile("" ::: "memory");

        // 16-step recurrence; z = C*f*h into Zbuf[n][t]
        #pragma unroll
        for (int t = 0; t < CHUNK; ++t) {
            float xt = Xbuf[w][pb][half][t];           // LDS broadcast
            h = clampf(__builtin_fmaf(r, h, Bv * xt)); // fma + med3 chain only
            g *= r;
            float f = __builtin_amdgcn_rcpf(1.0f + g); // v_rcp_f32, off chain
            Zbuf[w][half][n][t] = Cv * h * f;
        }
        __builtin_amdgcn_wave_barrier();
        asm volatile("" ::: "memory");

#if USE_WMMA_F32
        // Y[t] = sum_n Z[t][n] via 4x v_wmma_f32_16x16x4_f32, B = ones.
        v8f acc0 = {};
        v8f acc1 = {};
        const v2f ones = {1.0f, 1.0f};
        const int m = n;                               // A-matrix row = time t
        #pragma unroll
        for (int k = 0; k < 4; ++k) {
            const int kk = 4 * k + 2 * half;
            v2f a0; a0[0] = Zbuf[w][0][kk][m]; a0[1] = Zbuf[w][0][kk + 1][m];
            acc0 = __builtin_amdgcn_wmma_f32_16x16x4_f32(
                       false, a0, false, ones, (short)0, acc0, false, false);
            v2f a1; a1[0] = Zbuf[w][1][kk][m]; a1[1] = Zbuf[w][1][kk + 1][m];
            acc1 = __builtin_amdgcn_wmma_f32_16x16x4_f32(
                       false, a1, false, ones, (short)0, acc1, false, false);
        }
        // D rows: lane<16 -> t=0..7 in acc[i]; lane>=16 -> t=8..15.
        if (lane == 0 || lane == 16) {
            #pragma unroll
            for (int i = 0; i < 8; ++i) {
                int t = i + 8 * half;
                ob[(size_t)(l0base + c * CHUNK + t) * DIM + ch0] =
                    acc0[i] + Xbuf[w][pb][0][t] * Dd0;
            }
        }
        if (lane == 1 || lane == 17) {
            #pragma unroll
            for (int i = 0; i < 8; ++i) {
                int t = i + 8 * half;
                ob[(size_t)(l0base + c * CHUNK + t) * DIM + ch0 + 1] =
                    acc1[i] + Xbuf[w][pb][1][t] * Dd1;
            }
        }
#else
        float Y = 0.0f;
        #pragma unroll
        for (int k = 0; k < STATE; ++k) Y += Zbuf[w][half][k][n];
        const float Dd = half ? Dd1 : Dd0;
        ob[(size_t)(l0base + c * CHUNK + n) * DIM + ch] =
            Y + Xbuf[w][pb][half][n] * Dd;
#endif
        pb ^= 1;
    }
}

// ---------------- Launch -------------------------------------------------
extern "C" void kernel_launch(void* const* d_in, const int* in_sizes, int n_in,
                              void* d_out, int out_size, void* d_ws, size_t ws_size,
                              hipStream_t stream) {
    (void)in_sizes; (void)n_in; (void)out_size;
    const float* x  = (const float*)d_in[0];
    const float* A  = (const float*)d_in[1];
    const float* B  = (const float*)d_in[2];
    const float* C  = (const float*)d_in[3];
    const float* Dv = (const float*)d_in[4];
    float* out = (float*)d_out;
    float* E   = (float*)d_ws;

    const size_t needE =
        (size_t)(NSEG - 1) * BATCH * DIM * STATE * sizeof(float);  // ~0.9 MB
    const int nseg = (d_ws != nullptr && ws_size >= needE) ? NSEG : 1;

    dim3 block(WAVES * 32);
    if (nseg > 1) {
        ssm_carry_kernel<<<dim3(BLOCKS_PER_SEG * (NSEG - 1)), block, 0, stream>>>(
            x, A, B, E);
    }
    ssm_out_kernel<<<dim3(BLOCKS_PER_SEG * nseg), block, 0, stream>>>(
        x, A, B, C, Dv, E, out, SEQ / nseg);
}